// DINOv2LoRAFeaturizer_69372311765563
// MI455X (gfx1250) — compile-verified
//
#include <hip/hip_runtime.h>
#include <cstdint>
#include <cstddef>

// ---------------------------------------------------------------------------
// Log-bin featurizer for MI455X (gfx1250).
//   in : x [8, 37, 37, 384] f32  (channel innermost)
//   out: [8, 29*384, 37, 37] f32 (pixel innermost), bins 0..16 filled:
//        bins 0..8  : clamped shifts (dy,dx in {-1,0,1}) of x (transposed)
//        bins 9..16 : clamped shifts (dy,dx in {-3,0,3}\{0,0}) of 3x3 avg-pool
//        bins 17..28: zeros
// Memory-bound (~488 MB stores, 23.3 TB/s -> ~22 us floor).
//   - Gather-transpose staged into LDS by the Tensor Data Mover (TDM):
//     one 2D descriptor per wave (16B rows, 1536B stride), s_wait_tensorcnt.
//     Fallback: per-lane global_load_async_to_lds_b128 (ASYNCcnt path).
//   - All output written with non-temporal stores (output >> L2, write-once).
// ---------------------------------------------------------------------------

#define B_    8
#define W_    37            // w0 == h0
#define D_    384
#define NB    29
#define NFILL 17
#define PLANE (W_ * W_)     // 1369
#define CCH   4             // channels per block (one float4 / pixel)
#define NBLK  (B_ * (D_ / CCH))   // 768

typedef __attribute__((ext_vector_type(4))) unsigned int v4u;
typedef __attribute__((ext_vector_type(8))) int          v8i;
typedef __attribute__((ext_vector_type(4))) int          v4i;
typedef __attribute__((ext_vector_type(4))) float        fvec4;

__device__ __constant__ int c_dy[NFILL] =
    {-1,-1,-1, 0,0,0, 1,1,1,  -3,-3,-3, 0,0, 3,3,3};
__device__ __constant__ int c_dx[NFILL] =
    {-1, 0, 1,-1,0,1,-1,0,1,  -3, 0, 3,-3,3,-3,0,3};

__global__ __launch_bounds__(256)
void logbin_kernel(const float* __restrict__ x, float* __restrict__ out) {
    __shared__ float4 lx[PLANE];   // x[b, pixel, c0..c0+3]   (transposed stage)
    __shared__ float4 lp[PLANE];   // 3x3 avg pool of lx

    const int tid = threadIdx.x;
    const int b   = blockIdx.x / (D_ / CCH);
    const int c0  = (blockIdx.x % (D_ / CCH)) * CCH;

    const uint64_t gbase =
        (uint64_t)(uintptr_t)(x + ((size_t)b * PLANE) * D_ + c0);
    const uint32_t lbase = (uint32_t)(uintptr_t)&lx[0];

#if __has_builtin(__builtin_amdgcn_tensor_load_to_lds)
    // ---- Phase 1 (TDM): each wave issues one 2D tensor descriptor ----------
    // tile: tile_dim0 = 4 elems (16B rows), tile_dim1 = rows for this wave,
    // row stride in memory = 384 elems (1536B); lands contiguously in LDS.
    {
        const int wv   = tid >> 5;                 // wave id, 8 waves
        const int r0   = wv * 172;                 // ceil(1369/8) = 172
        const int rows = (PLANE - r0 < 172) ? (PLANE - r0) : 172;
        const uint64_t ga = gbase + (uint64_t)r0 * (D_ * 4);
        const uint32_t la = lbase + (uint32_t)r0 * 16u;

        v4u g0;
        g0.x = 1u;                                  // count=1 (valid descriptor)
        g0.y = la;                                  // lds_addr (bytes)
        g0.z = (uint32_t)(ga & 0xFFFFFFFFu);        // global_addr[31:0]
        g0.w = (uint32_t)((ga >> 32) & 0x01FFFFFFu) // global_addr[56:32]
               | 0x80000000u;                       // type=2 ("image")

        v8i g1 = { (int)(2u << 16),                 // data_size=2 -> 4 bytes
                   (int)(4u << 16),                 // tensor_dim0 = 4 (bits 63:48)
                   (int)((unsigned)PLANE << 16),    // tensor_dim1 lo (bits 95:80)
                   (int)(4u << 16),                 // tile_dim0 = 4 (bits 127:112)
                   rows,                            // tile_dim1 (bits 143:128)
                   D_,                              // tensor_dim0_stride = 384
                   0, 0 };                          // stride hi / dim1_stride
        v4i gz4 = { 0, 0, 0, 0 };
#if defined(__clang_major__) && (__clang_major__ >= 23)
        v8i gz8 = { 0, 0, 0, 0, 0, 0, 0, 0 };
        __builtin_amdgcn_tensor_load_to_lds(g0, g1, gz4, gz4, gz8, 0);
#else
        __builtin_amdgcn_tensor_load_to_lds(g0, g1, gz4, gz4, 0);
#endif
        __builtin_amdgcn_s_wait_tensorcnt(0);
    }
    __syncthreads();
#else
    // ---- Phase 1 (fallback): per-lane async global->LDS b128 gather --------
    for (int i = tid; i < PLANE; i += 256) {
        uint64_t ga = gbase + (uint64_t)i * (D_ * 4);
        uint32_t la = lbase + (uint32_t)i * 16u;
        asm volatile("global_load_async_to_lds_b128 %0, %1, off"
                     :: "v"(la), "v"(ga) : "memory");
    }
    asm volatile("s_wait_asynccnt 0" ::: "memory");
    __syncthreads();
#endif

    // ---- Phase 2: 3x3 avg pool, count_include_pad=False --------------------
    for (int i = tid; i < PLANE; i += 256) {
        int y  = i / W_;
        int xx = i - y * W_;
        int ys = max(y  - 1, 0), ye = min(y  + 1, W_ - 1);
        int xs = max(xx - 1, 0), xe = min(xx + 1, W_ - 1);
        float4 s = make_float4(0.f, 0.f, 0.f, 0.f);
        for (int yy = ys; yy <= ye; ++yy)
            for (int xw = xs; xw <= xe; ++xw) {
                float4 v = lx[yy * W_ + xw];
                s.x += v.x; s.y += v.y; s.z += v.z; s.w += v.w;
            }
        float cnt = (float)((ye - ys + 1) * (xe - xs + 1));
        lp[i] = make_float4(s.x / cnt, s.y / cnt, s.z / cnt, s.w / cnt);
    }
    __syncthreads();

    // ---- Phase 3: emit 17 bins x 4 channels, coalesced NT stores -----------
    float* outb = out + (size_t)b * ((size_t)NB * D_ * PLANE)
                      + (size_t)c0 * PLANE;
    for (int i = tid; i < PLANE; i += 256) {
        int y  = i / W_;
        int xx = i - y * W_;
        #pragma unroll 1
        for (int bin = 0; bin < NFILL; ++bin) {
            int sy = min(max(y  + c_dy[bin], 0), W_ - 1);
            int sx = min(max(xx + c_dx[bin], 0), W_ - 1);
            const float4* src = (bin < 9) ? lx : lp;
            float4 v = src[sy * W_ + sx];
            float* op = outb + (size_t)bin * ((size_t)D_ * PLANE) + i;
            __builtin_nontemporal_store(v.x, op + 0 * PLANE);
            __builtin_nontemporal_store(v.y, op + 1 * PLANE);
            __builtin_nontemporal_store(v.z, op + 2 * PLANE);
            __builtin_nontemporal_store(v.w, op + 3 * PLANE);
        }
    }
}

// ---- Zero-fill bins 17..28 (contiguous 12*384*1369 floats per batch) -------
__global__ __launch_bounds__(256)
void zero_tail_kernel(float* __restrict__ out) {
    const int perB    = (12 * D_ * PLANE) / 4;       // 1,577,088 float4
    const int strideB = (NB * D_ * PLANE) / 4;       // 3,811,296 float4
    const int startB  = (NFILL * D_ * PLANE) / 4;    // 2,234,208 float4
    const long total  = (long)B_ * perB;
    fvec4* out4 = (fvec4*)out;
    const fvec4 z = { 0.f, 0.f, 0.f, 0.f };
    for (long q = (long)blockIdx.x * blockDim.x + threadIdx.x;
         q < total;
         q += (long)gridDim.x * blockDim.x) {
        int b = (int)(q / perB);
        int r = (int)(q - (long)b * perB);
        __builtin_nontemporal_store(z, out4 + (size_t)b * strideB + startB + r);
    }
}

extern "C" void kernel_launch(void* const* d_in, const int* in_sizes, int n_in,
                              void* d_out, int out_size, void* d_ws, size_t ws_size,
                              hipStream_t stream) {
    (void)in_sizes; (void)n_in; (void)out_size; (void)d_ws; (void)ws_size;
    const float* x = (const float*)d_in[0];
    float* out = (float*)d_out;

    logbin_kernel<<<NBLK, 256, 0, stream>>>(x, out);
    zero_tail_kernel<<<2048, 256, 0, stream>>>(out);
}